// MeanSquaredError3D2_1906965480050
// MI455X (gfx1250) — compile-verified
//
#include <hip/hip_runtime.h>
#include <math.h>

#define NJ   24
#define COL  14
#define CC   196     // COL*COL
#define CCC  2744    // COL^3
#define TMPW 3       // gaussian half-window

typedef __attribute__((ext_vector_type(2))) float v2f;
typedef __attribute__((ext_vector_type(8))) float v8f;

// Sum 64 f32 values (2 per lane, wave32) with one V_WMMA_F32_16X16X4_F32:
// D = A x Ones + 0  ->  D[m][*] = rowsum(m).  C/D layout: lanes 0-15 hold
// rows 0-7 in VGPRs 0-7, lanes 16-31 hold rows 8-15, so summing the 8 D
// registers and xor-shuffling across lane 16 yields the full 64-slot sum.
__device__ __forceinline__ float wave_sum_wmma(float a0, float a1) {
    v2f A;  A[0] = a0;   A[1] = a1;
    v2f Bn; Bn[0] = 1.f; Bn[1] = 1.f;     // all-ones B: layout-independent
    v8f C = {};
    C = __builtin_amdgcn_wmma_f32_16x16x4_f32(false, A, false, Bn,
                                              (short)0, C, false, false);
    float s = C[0] + C[1] + C[2] + C[3] + C[4] + C[5] + C[6] + C[7];
    s += __shfl_xor(s, 16, 32);           // combine row halves
    return s;                              // wave total, broadcast to all lanes
}

// Block (256 threads = 8 waves) sum of 2 accumulators per thread.
__device__ __forceinline__ float block_sum(float a0, float a1, float* lds8) {
    float s = wave_sum_wmma(a0, a1);
    if ((threadIdx.x & 31) == 0) lds8[threadIdx.x >> 5] = s;
    __syncthreads();
    float t = lds8[0] + lds8[1] + lds8[2] + lds8[3]
            + lds8[4] + lds8[5] + lds8[6] + lds8[7];
    __syncthreads();
    return t;
}

__global__ __launch_bounds__(256)
void joint_kernel(const float* __restrict__ o,   const float* __restrict__ h,
                  const float* __restrict__ o3D, const float* __restrict__ h3D,
                  const float* __restrict__ t,   const float* __restrict__ t3D,
                  const float* __restrict__ v,
                  float* __restrict__ ws, int NB) {
    __shared__ float sv[256];
    __shared__ int   si[256];
    __shared__ float lds8[8];

    const int tid = threadIdx.x;
    const int blk = blockIdx.x;
    const int b = blk / NJ, j = blk % NJ;
    const size_t bj = (size_t)b * NJ + j;

    // ---- per-joint scalars (uniform across block) ----
    const float tx  = t[bj * 2 + 0],  ty  = t[bj * 2 + 1];
    const float t3x = t3D[bj * 3 + 0], t3y = t3D[bj * 3 + 1], t3z = t3D[bj * 3 + 2];
    const float v0  = v[bj * 3 + 0];

    const int mux = (int)floorf(tx * COL + 0.5f);
    const int muy = (int)floorf(ty * COL + 0.5f);
    const int m3x = (int)floorf(t3x * COL + 0.5f);
    const int m3y = (int)floorf(t3y * COL + 0.5f);
    const int m3z = (int)floorf(t3z * COL + 7.0f + 0.5f);

    const bool vis = (v0 == 1.0f);
    const bool valid2d = !((mux - TMPW >= COL) || (muy - TMPW >= COL) ||
                           (mux + TMPW + 1 <= 0) || (muy + TMPW + 1 <= 0));
    const bool valid3d = !((m3x - TMPW >= COL) || (m3y - TMPW >= COL) ||
                           (m3z - TMPW >= COL) ||
                           (m3x + TMPW + 1 < 0) || (m3y + TMPW + 1 < 0) ||
                           (m3z + TMPW + 1 < 0));
    const bool jmask = vis && valid2d && valid3d;

    if (!jmask) {            // uniform early-out: joint contributes nothing
        if (tid == 0) {
            ws[0 * NB + blk] = 0.f; ws[1 * NB + blk] = 0.f;
            ws[2 * NB + blk] = 0.f; ws[3 * NB + blk] = 0.f;
            ws[4 * NB + blk] = 0.f;
        }
        return;
    }

    // ---- 2D heatmap: 196 elems = 49 float4, threads 0..48 ----
    float acc2 = 0.f;
    float best2 = -INFINITY; int bi2 = 1 << 30;
    const float4* h2 = (const float4*)(h + bj * CC);
    if (tid < CC / 4) {
        float4 hv = h2[tid];
        float vals[4] = {hv.x, hv.y, hv.z, hv.w};
        #pragma unroll
        for (int c = 0; c < 4; ++c) {
            int idx = tid * 4 + c;
            int y = idx / COL, x = idx - y * COL;
            int dx = x - mux, dy = y - muy;
            float g = 0.f;
            if (dx >= -TMPW && dx <= TMPW && dy >= -TMPW && dy <= TMPW)
                g = expf(-0.5f * (float)(dx * dx + dy * dy));
            float d = vals[c] - g;
            acc2 += d * d;
            if (vals[c] > best2) { best2 = vals[c]; bi2 = idx; }
        }
    }

    // ---- 3D heatmap: 2744 elems = 686 float4, coalesced b128 stream ----
    float acc0 = 0.f, acc1 = 0.f;
    float best3 = -INFINITY; int bi3 = 1 << 30;
    const float4* h3 = (const float4*)(h3D + bj * CCC);
    #pragma unroll
    for (int k = 0; k < 3; ++k) {
        int q = tid + k * 256;
        if (q < CCC / 4) {
            float4 hv = h3[q];
            float vals[4] = {hv.x, hv.y, hv.z, hv.w};
            #pragma unroll
            for (int c = 0; c < 4; ++c) {
                int idx = q * 4 + c;
                int z = idx / CC, r = idx - z * CC;
                int y = r / COL,  x = r - y * COL;
                int dx = x - m3x, dy = y - m3y, dz = z - m3z;
                float g = 0.f;
                if (dx >= -TMPW && dx <= TMPW && dy >= -TMPW && dy <= TMPW &&
                    dz >= -TMPW && dz <= TMPW)
                    g = expf(-0.5f * (float)(dx * dx + dy * dy + dz * dz));
                float d = vals[c] - g;
                if (c & 1) acc1 += d * d; else acc0 += d * d;
                if (vals[c] > best3) { best3 = vals[c]; bi3 = idx; }
            }
        }
    }

    // ---- block sums (WMMA-based) ----
    float s1 = block_sum(acc2, 0.f, lds8);   // sum (h  - tt )^2
    float s3 = block_sum(acc0, acc1, lds8);  // sum (h3 - tt3)^2

    // ---- 2D argmax (first-index tie-break) ----
    sv[tid] = best2; si[tid] = bi2;
    __syncthreads();
    for (int off = 128; off > 0; off >>= 1) {
        if (tid < off) {
            float ov = sv[tid + off]; int oi = si[tid + off];
            if (ov > sv[tid] || (ov == sv[tid] && oi < si[tid])) {
                sv[tid] = ov; si[tid] = oi;
            }
        }
        __syncthreads();
    }
    int am2 = si[0];
    __syncthreads();

    // ---- 3D argmax ----
    sv[tid] = best3; si[tid] = bi3;
    __syncthreads();
    for (int off = 128; off > 0; off >>= 1) {
        if (tid < off) {
            float ov = sv[tid + off]; int oi = si[tid + off];
            if (ov > sv[tid] || (ov == sv[tid] && oi < si[tid])) {
                sv[tid] = ov; si[tid] = oi;
            }
        }
        __syncthreads();
    }
    int am3 = si[0];

    if (tid == 0) {
        const float sc = 1.0f / COL;
        // 2D decode + offset gather
        int y2 = am2 / COL, x2 = am2 - y2 * COL;
        float gx = o[((size_t)b * 2 * NJ + j) * CC + am2];
        float gy = o[((size_t)b * 2 * NJ + NJ + j) * CC + am2];
        float ex = gx + (float)x2 * sc - tx;
        float ey = gy + (float)y2 * sc - ty;
        float s2 = ex * ex + ey * ey;
        // 3D decode + offset gather
        int z3 = am3 / CC; int r3 = am3 - z3 * CC;
        int y3 = r3 / COL, x3 = r3 - y3 * COL;
        size_t ob = ((size_t)b * 3 * NJ + j) * (size_t)CCC + am3;
        float g0 = o3D[ob];
        float g1 = o3D[ob + (size_t)NJ * CCC];
        float g2 = o3D[ob + (size_t)2 * NJ * CCC];
        float e0 = g0 + (float)x3 * sc - t3x;
        float e1 = g1 + (float)y3 * sc - t3y;
        float e2 = g2 + (float)(z3 - 7) * sc - t3z;
        float s4 = e0 * e0 + e1 * e1 + e2 * e2;

        ws[0 * NB + blk] = s1;
        ws[1 * NB + blk] = s2;
        ws[2 * NB + blk] = s3;
        ws[3 * NB + blk] = s4;
        ws[4 * NB + blk] = 1.0f;    // joint count
    }
}

__global__ __launch_bounds__(256)
void finish_kernel(const float* __restrict__ ws, float* __restrict__ out, int NB) {
    __shared__ float lds8[8];
    const int tid = threadIdx.x;
    float a[5] = {0.f, 0.f, 0.f, 0.f, 0.f};
    for (int r = tid; r < NB; r += 256) {
        #pragma unroll
        for (int c = 0; c < 5; ++c) a[c] += ws[(size_t)c * NB + r];
    }
    float tot[5];
    #pragma unroll
    for (int c = 0; c < 5; ++c) tot[c] = block_sum(a[c], 0.f, lds8);
    if (tid == 0) {
        float cnt = tot[4];          // Nv == cnt in this loss
        out[0] = (sqrtf(tot[0]) + sqrtf(tot[1]) +
                  sqrtf(tot[2]) + sqrtf(tot[3])) / cnt;
    }
}

extern "C" void kernel_launch(void* const* d_in, const int* in_sizes, int n_in,
                              void* d_out, int out_size, void* d_ws, size_t ws_size,
                              hipStream_t stream) {
    const float* o   = (const float*)d_in[0];
    const float* h   = (const float*)d_in[1];
    const float* o3D = (const float*)d_in[2];
    const float* h3D = (const float*)d_in[3];
    const float* t   = (const float*)d_in[4];
    const float* t3D = (const float*)d_in[5];
    const float* v   = (const float*)d_in[6];

    const int B  = in_sizes[1] / (NJ * CC);
    const int NB = B * NJ;
    float* wsf = (float*)d_ws;   // needs NB*5 floats (120 KB for B=256)

    joint_kernel<<<NB, 256, 0, stream>>>(o, h, o3D, h3D, t, t3D, v, wsf, NB);
    finish_kernel<<<1, 256, 0, stream>>>(wsf, (float*)d_out, NB);
}